// CapsShapeLayer_52544629899592
// MI455X (gfx1250) — compile-verified
//
#include <hip/hip_runtime.h>

// CapsNet dynamic routing, fully fused for MI455X (gfx1250, wave32).
//
// Roofline: fused kernel touches only x (9.4 MB) + W (164 KB, L2-resident) +
// out (160 KB) => ~10 MB @ 23.3 TB/s ~= 0.4 us floor, vs ~750 MB (~32 us) if
// u_hat (188 MB) were materialized. u_hat is *never* formed:
//   s[d,o]    = sum_m ( sum_p c[d,m,p]*x[m,p,:] ) . W[d,m,o,:]
//   db[d,m,p] = x[m,p,:] . ( sum_o W[d,m,o,:]*v[d,o] )
// leaving two per-m fp32 GEMM families executed with V_WMMA_F32_16X16X4_F32.
//
// WMMA operand loads/stores are BRANCHLESS via LDS padding (pad-row garbage
// only lands in unstored output rows/cols or in blog's never-read pad region).
// The per-wave m index is readfirstlane'd so loop control is scalar, and the
// 4-m loop is fully unrolled to give 4 independent WMMA accumulation chains
// per wave (hides WMMA + LDS latency).

typedef __attribute__((ext_vector_type(2))) float v2f;
typedef __attribute__((ext_vector_type(8))) float v8f;

#define Bn 256
#define Dn 10
#define Mn 32
#define Pn 36
#define In 8
#define On 16
#define DP 16      // D padded to WMMA tile
#define PP 48      // P padded to 3x16 row tiles
#define IP 16      // I padded to WMMA tile (xc store stride)
#define EPSf 1e-7f

// LDS float offsets
#define XS_OFF 0                       // xs  [Mn][PP][In]   12288
#define BL_OFF 12288                   // blog[DP][Mn][PP]   24576
#define CC_OFF 36864                   // cc  [DP][Mn][Pn]   18432
#define XC_OFF 55296                   // xc  [DP][Mn][IP]    8192
#define WV_OFF 63488                   // wv  [DP][Mn][In]    4096
#define SV_OFF 67584                   // s   [Dn][On]         160
#define VV_OFF 67744                   // v   [Dn][On]         160
#define LDS_FLOATS 67904               // 271,616 B (< 320 KB/WGP)

__global__ __launch_bounds__(256)
void caps_routing_fused(const float* __restrict__ x,
                        const float* __restrict__ W,
                        float* __restrict__ out)
{
    extern __shared__ float lds[];
    float* xs   = lds + XS_OFF;
    float* blog = lds + BL_OFF;
    float* cc   = lds + CC_OFF;
    float* xc   = lds + XC_OFF;
    float* wv   = lds + WV_OFF;
    float* sv   = lds + SV_OFF;
    float* vv   = lds + VV_OFF;

    const int tid  = threadIdx.x;
    const int lane = tid & 31;
    // wave index is wave-uniform: tell the compiler so loop control goes SALU
    const int wave = __builtin_amdgcn_readfirstlane(tid >> 5);
    const int nlo  = lane & 15;               // lane mod 16
    const int kb   = (lane < 16) ? 0 : 2;     // K sub-offset for this half-wave
    const int rhi  = (lane < 16) ? 0 : 8;     // row offset for C/D halves
    const int b    = blockIdx.x;

    // ---- stage x[b] into padded LDS (coalesced); zero padded logits ----
    const float* xb = x + (size_t)b * (Mn * Pn * In);
    for (int j = tid; j < Mn * Pn * In; j += 256) {
        const int m = j / (Pn * In), r = j % (Pn * In);
        xs[m * (PP * In) + r] = xb[j];        // pad rows p=36..47 left as-is
    }
    for (int j = tid; j < DP * Mn * PP; j += 256) blog[j] = 0.0f;
    __syncthreads();

    for (int it = 0; it < 3; ++it) {
        // ================= routing weights -> cc =================
        if (it == 0) {
            // softmax over D of zeros == exactly 1/D
            for (int j = tid; j < Dn * Mn * Pn; j += 256) cc[j] = 0.1f;
        } else if (it == 1) {
            // softmax over D for each (m,p)
            for (int j = tid; j < Mn * Pn; j += 256) {
                const int m = j / Pn, p = j % Pn;
                const int base = m * PP + p;
                float mx = -3.4e38f;
                for (int d = 0; d < Dn; ++d)
                    mx = fmaxf(mx, blog[d * (Mn * PP) + base]);
                float sum = 0.0f;
                for (int d = 0; d < Dn; ++d)
                    sum += __expf(blog[d * (Mn * PP) + base] - mx);
                const float inv = 1.0f / sum;
                for (int d = 0; d < Dn; ++d)
                    cc[d * (Mn * Pn) + m * Pn + p] =
                        __expf(blog[d * (Mn * PP) + base] - mx) * inv;
            }
        } else {
            // final: softmax over P for each (d,m)
            for (int j = tid; j < Dn * Mn; j += 256) {
                const int d = j / Mn, m = j % Mn;
                const float* bl = &blog[(d * Mn + m) * PP];
                float mx = -3.4e38f;
                for (int p = 0; p < Pn; ++p) mx = fmaxf(mx, bl[p]);
                float sum = 0.0f;
                for (int p = 0; p < Pn; ++p) sum += __expf(bl[p] - mx);
                const float inv = 1.0f / sum;
                float* cd = &cc[(d * Mn + m) * Pn];
                for (int p = 0; p < Pn; ++p) cd[p] = __expf(bl[p] - mx) * inv;
            }
        }
        __syncthreads();

        // ====== xc[d,m,i] = sum_p cc[d,m,p] * xs[m,p,i]  (WMMA, branchless) ======
        // Per m: A = cc (16x36, d-padded), B = xs (36x16, i over-read into pads),
        // K = 36 in 9 steps of V_WMMA_F32_16X16X4_F32.
        // 4 m's fully unrolled -> 4 independent accumulation chains per wave.
        {
            v8f acc[4] = {{}, {}, {}, {}};
            #pragma unroll
            for (int k0 = 0; k0 < Pn; k0 += 4) {
                const int ka = k0 + kb;
                #pragma unroll
                for (int mm = 0; mm < 4; ++mm) {
                    const int m = wave * 4 + mm;
                    const float* am = &cc[nlo * (Mn * Pn) + m * Pn];
                    const float* bmbase = &xs[m * (PP * In)];
                    v2f a, bv;
                    a.x  = am[ka + 0];
                    a.y  = am[ka + 1];
                    bv.x = bmbase[(ka + 0) * In + nlo];  // nlo>=8: pad cols
                    bv.y = bmbase[(ka + 1) * In + nlo];
                    acc[mm] = __builtin_amdgcn_wmma_f32_16x16x4_f32(
                        false, a, false, bv, (short)0, acc[mm], false, false);
                }
            }
            // unconditional 16x16 stores into padded xc
            #pragma unroll
            for (int mm = 0; mm < 4; ++mm) {
                const int m = wave * 4 + mm;
                #pragma unroll
                for (int r = 0; r < 8; ++r)
                    xc[(r + rhi) * (Mn * IP) + m * IP + nlo] = acc[mm][r];
            }
        }
        __syncthreads();

        // ====== s[d,o] = sum_{m,i} W[d,m,o,i] * xc[d,m,i]  (VALU, K=256) ======
        if (tid < Dn * On) {
            const int d = tid / On, o = tid % On;
            float acc = 0.0f;
            const float* wp = W + ((size_t)d * Mn * On + o) * In;
            for (int m = 0; m < Mn; ++m) {
                const float* wm = wp + (size_t)m * On * In;
                const float* xm = &xc[d * (Mn * IP) + m * IP];
                #pragma unroll
                for (int i = 0; i < In; ++i) acc += wm[i] * xm[i];
            }
            sv[tid] = acc;
        }
        __syncthreads();

        // ====== v = squash(s) ======
        if (tid < Dn * On) {
            const int d = tid / On;
            float sq = 0.0f;
            #pragma unroll
            for (int o = 0; o < On; ++o) { const float t = sv[d * On + o]; sq += t * t; }
            const float scale = (sq / (1.0f + sq)) * rsqrtf(sq + EPSf);
            vv[tid] = sv[tid] * scale;
        }
        __syncthreads();

        if (it == 2) {
            if (tid < Dn * On)
                out[(size_t)b * Dn * On + tid] = vv[tid];
            break;
        }

        // ====== wv[d,m,i] = sum_o W[d,m,o,i] * v[d,o]  (VALU, K=16) ======
        for (int j = tid; j < Dn * Mn * In; j += 256) {
            const int d = j / (Mn * In);
            const int mi = j % (Mn * In);
            const int m = mi / In, i = mi % In;
            float acc = 0.0f;
            const float* wm = W + ((size_t)d * Mn + m) * On * In + i;
            #pragma unroll
            for (int o = 0; o < On; ++o) acc += wm[o * In] * vv[d * On + o];
            wv[d * (Mn * In) + mi] = acc;
        }
        __syncthreads();

        // ====== blog[d,m,p] += sum_i xs[m,p,i] * wv[d,m,i]  (WMMA, branchless) ======
        // Per m: 3 row tiles (P padded to 48), N = D padded to 16, K = 8 in 2
        // steps. 4 m's x 3 tiles = 12 independent chains; pad-row results land
        // in blog's never-read pad region.
        #pragma unroll
        for (int mm = 0; mm < 4; ++mm) {
            const int m = wave * 4 + mm;
            const float* amb = &xs[m * (PP * In)];
            const float* bmb = &wv[nlo * (Mn * In) + m * In];
            #pragma unroll
            for (int t = 0; t < 3; ++t) {
                v8f acc = {};
                const int prow = 16 * t + nlo;        // 0..47, in padded range
                #pragma unroll
                for (int k0 = 0; k0 < In; k0 += 4) {
                    const int ka = k0 + kb;
                    v2f a, bv;
                    a.x  = amb[prow * In + ka + 0];
                    a.y  = amb[prow * In + ka + 1];
                    bv.x = bmb[ka + 0];               // nlo>=10 rows: pad garbage
                    bv.y = bmb[ka + 1];
                    acc = __builtin_amdgcn_wmma_f32_16x16x4_f32(
                        false, a, false, bv, (short)0, acc, false, false);
                }
                float* bl = &blog[nlo * (Mn * PP) + m * PP + 16 * t + rhi];
                #pragma unroll
                for (int r = 0; r < 8; ++r)
                    bl[r] += acc[r];                  // unconditional RMW in pads
            }
        }
        __syncthreads();
    }
}

extern "C" void kernel_launch(void* const* d_in, const int* in_sizes, int n_in,
                              void* d_out, int out_size, void* d_ws, size_t ws_size,
                              hipStream_t stream) {
    (void)in_sizes; (void)n_in; (void)out_size; (void)d_ws; (void)ws_size;
    const float* x = (const float*)d_in[0];   // (B, M, P, I) fp32
    const float* W = (const float*)d_in[1];   // (1, D, M, 1, O, I) fp32
    float* out = (float*)d_out;               // (B, D, O) fp32
    const int shmem = LDS_FLOATS * (int)sizeof(float);  // 271,616 B
    (void)hipFuncSetAttribute((const void*)caps_routing_fused,
                              hipFuncAttributeMaxDynamicSharedMemorySize, shmem);
    caps_routing_fused<<<Bn, 256, shmem, stream>>>(x, W, out);
}